// MACUnit_9844065042835
// MI455X (gfx1250) — compile-verified
//
#include <hip/hip_runtime.h>
#include <hip/hip_bf16.h>
#include <math.h>

// ---------------------------------------------------------------------------
// MAC recurrence, restructured:
//   cq   = [ctrl|question] @ Wcq.T + bcq                    (GEMM, wmma f32)
//   aw_s = <cq*Wca, context[b,s,:]>, softmax_s, new_ctrl    (reduction kernel)
//   memp = mem @ Wm.T + bm                                  (GEMM)
//   pq   = (new_ctrl*Wra) @ Wrc    [B,1024]                 (GEMM, fused z)
//   u    = memp*p + q ; ra_n = <u, K[b,:,n]>, softmax_n     (reduction kernel)
//   read = K[b,:,:] @ ra
//   mem' = [read|mem] @ Wwc.T + bwc                         (GEMM, fused with
//   cq'  = [new_ctrl|question] @ Wcq.T + bcq                 next-step cq)
// bca/brc/bra cancel inside softmax and are dropped exactly.
// ---------------------------------------------------------------------------

typedef __attribute__((ext_vector_type(2))) float v2f;
typedef __attribute__((ext_vector_type(8))) float v8f;

#define DIMD 512
#define BATCH 256
#define SEQ 32
#define NK 196
#define TM 16
#define TN 64
#define TK 64

struct GemmDesc {
  const float* X;      // A[:, 0:512]      (row-major, stride 512)
  const float* Y;      // A[:, 512:1024]   (nullable if K==512)
  const float* scale;  // optional per-k scale on A (length K)
  const float* W;
  const float* bias;   // nullable
  float* out;          // [256, N]
  int K;               // 512 or 1024 (multiple of 64)
  int N;               // multiple of 64
  int wT;              // 1: B[k][n] = W[n*K+k]   0: B[k][n] = W[k*N+n]
};

// out = A @ B + bias, M=256 fixed. Block = 128 threads (4 waves);
// each wave owns a 16x16 tile -> block covers 16(M) x 64(N).
__global__ __launch_bounds__(128)
void gemm_wmma_f32(GemmDesc d0, GemmDesc d1) {
  GemmDesc d = (blockIdx.z == 0) ? d0 : d1;
  const int n0 = blockIdx.x * TN;
  if (n0 >= d.N) return;                 // uniform per block (z-fused grids)
  const int m0 = blockIdx.y * TM;

  __shared__ float aT[TM][TK + 4];
  __shared__ float bT[TK][TN + 4];

  const int tid   = threadIdx.x;
  const int lane  = tid & 31;
  const int wave  = tid >> 5;
  const int mfrag = lane & 15;           // A row / B col within tile
  const int hi    = lane >> 4;           // lane half
  const int nOff  = wave * 16;

  v8f acc = {0.f, 0.f, 0.f, 0.f, 0.f, 0.f, 0.f, 0.f};

  for (int kk = 0; kk < d.K; kk += TK) {
    // ---- stage A tile: 16 x 64 (8 floats / thread, float4 loads) ----
    {
      int idx = tid * 8;
      int r = idx >> 6;
      int c = idx & 63;
      int kg = kk + c;                   // c..c+7 stay inside one 64-block
      const float* src = (kg < DIMD) ? (d.X + (size_t)(m0 + r) * DIMD + kg)
                                     : (d.Y + (size_t)(m0 + r) * DIMD + (kg - DIMD));
      float4 f0 = reinterpret_cast<const float4*>(src)[0];
      float4 f1 = reinterpret_cast<const float4*>(src)[1];
      float v[8] = {f0.x, f0.y, f0.z, f0.w, f1.x, f1.y, f1.z, f1.w};
      if (d.scale) {
#pragma unroll
        for (int j = 0; j < 8; ++j) v[j] *= d.scale[kg + j];
      }
#pragma unroll
      for (int j = 0; j < 8; ++j) aT[r][c + j] = v[j];
    }
    // ---- stage B tile: 64(k) x 64(n) ----
    if (d.wT) {
#pragma unroll
      for (int j = 0; j < 8; ++j) {
        int linear = tid + j * 128;          // 0..1023
        int n = linear & 63;
        int k = (linear >> 6) * 4;           // contiguous along k in memory
        float4 f = reinterpret_cast<const float4*>(
            d.W + (size_t)(n0 + n) * d.K + kk + k)[0];
        bT[k + 0][n] = f.x; bT[k + 1][n] = f.y;
        bT[k + 2][n] = f.z; bT[k + 3][n] = f.w;
      }
    } else {
#pragma unroll
      for (int j = 0; j < 8; ++j) {
        int linear = tid + j * 128;
        int c4 = linear & 15;                // float4 column
        int k  = linear >> 4;                // 0..63
        float4 f = reinterpret_cast<const float4*>(
            d.W + (size_t)(kk + k) * d.N + n0 + c4 * 4)[0];
        int n = c4 * 4;
        bT[k][n + 0] = f.x; bT[k][n + 1] = f.y;
        bT[k][n + 2] = f.z; bT[k][n + 3] = f.w;
      }
    }
    __syncthreads();

    // ---- 16 WMMAs over this K-chunk ----
    // A frag (16x4 f32): lane holds A[m= lane&15][k = 2*(lane>>4)+j], j=0..1
    // B frag (4x16 f32): lane holds B[k = 2*(lane>>4)+j][n = lane&15]
#pragma unroll
    for (int kb = 0; kb < TK; kb += 4) {
      v2f a, b;
      a[0] = aT[mfrag][kb + 2 * hi + 0];
      a[1] = aT[mfrag][kb + 2 * hi + 1];
      b[0] = bT[kb + 2 * hi + 0][nOff + mfrag];
      b[1] = bT[kb + 2 * hi + 1][nOff + mfrag];
      acc = __builtin_amdgcn_wmma_f32_16x16x4_f32(
          false, a, false, b, (short)0, acc, false, false);
    }
    __syncthreads();
  }

  // C/D layout: acc[j] -> (M = j + 8*hi, N = lane&15)
  const int n = n0 + nOff + mfrag;
  float bv = d.bias ? d.bias[n] : 0.f;
#pragma unroll
  for (int j = 0; j < 8; ++j) {
    int m = m0 + j + 8 * hi;
    d.out[(size_t)m * d.N + n] = acc[j] + bv;
  }
}

// ---- control attention: softmax over S=32, one block per batch ----
__global__ __launch_bounds__(256)
void control_attn(const float* __restrict__ cq, const float* __restrict__ context,
                  const float* __restrict__ Wca, float* __restrict__ new_ctrl) {
  const int b = blockIdx.x;
  __shared__ float cqw[DIMD];
  __shared__ float awS[SEQ];
  __shared__ float aS[SEQ];
  const int tid = threadIdx.x;
  for (int e = tid; e < DIMD; e += 256) cqw[e] = cq[(size_t)b * DIMD + e] * Wca[e];
  __syncthreads();
  const int lane = tid & 31, wave = tid >> 5;
  const float* ctx = context + (size_t)b * SEQ * DIMD;
  for (int s = wave; s < SEQ; s += 8) {
    float acc = 0.f;
    for (int e = lane; e < DIMD; e += 32) acc += cqw[e] * ctx[(size_t)s * DIMD + e];
#pragma unroll
    for (int off = 16; off > 0; off >>= 1) acc += __shfl_down(acc, off, 32);
    if (lane == 0) awS[s] = acc;   // +bca omitted: constant over s, cancels
  }
  __syncthreads();
  if (tid < 32) {
    float x = awS[tid];
    float m = x;
#pragma unroll
    for (int off = 16; off > 0; off >>= 1) m = fmaxf(m, __shfl_xor(m, off, 32));
    float e = expf(x - m);
    float sum = e;
#pragma unroll
    for (int off = 16; off > 0; off >>= 1) sum += __shfl_xor(sum, off, 32);
    aS[tid] = e / sum;
  }
  __syncthreads();
  for (int e = tid; e < DIMD; e += 256) {
    float acc = 0.f;
#pragma unroll 8
    for (int s = 0; s < SEQ; ++s) acc += aS[s] * ctx[(size_t)s * DIMD + e];
    new_ctrl[(size_t)b * DIMD + e] = acc;
  }
}

// ---- read attention: u = memp*p+q ; softmax_n <u,K> ; read = K @ a ----
__global__ __launch_bounds__(256)
void read_attn(const float* __restrict__ memp, const float* __restrict__ pq,
               const float* __restrict__ knowledge, float* __restrict__ readv) {
  const int b = blockIdx.x;
  const int tid = threadIdx.x;
  __shared__ float u[DIMD];
  __shared__ float red[256];
  __shared__ float attn[NK];
  for (int k = tid; k < DIMD; k += 256)
    u[k] = memp[(size_t)b * DIMD + k] * pq[(size_t)b * 1024 + k]
         + pq[(size_t)b * 1024 + DIMD + k];
  __syncthreads();
  const float* know = knowledge + (size_t)b * DIMD * NK;
  float raw = 0.f;
  if (tid < NK) {
    for (int k = 0; k < DIMD; ++k) raw += u[k] * know[(size_t)k * NK + tid];
  }
  red[tid] = (tid < NK) ? raw : -3.4e38f;
  __syncthreads();
  for (int s = 128; s > 0; s >>= 1) {
    if (tid < s) red[tid] = fmaxf(red[tid], red[tid + s]);
    __syncthreads();
  }
  float m = red[0];
  __syncthreads();
  float ex = (tid < NK) ? expf(raw - m) : 0.f;
  red[tid] = ex;
  __syncthreads();
  for (int s = 128; s > 0; s >>= 1) {
    if (tid < s) red[tid] += red[tid + s];
    __syncthreads();
  }
  float inv = 1.f / red[0];
  if (tid < NK) attn[tid] = ex * inv;
  __syncthreads();
  const int lane = tid & 31, wave = tid >> 5;
  for (int e = wave; e < DIMD; e += 8) {
    float acc = 0.f;
    for (int n = lane; n < NK; n += 32) acc += attn[n] * know[(size_t)e * NK + n];
#pragma unroll
    for (int off = 16; off > 0; off >>= 1) acc += __shfl_down(acc, off, 32);
    if (lane == 0) readv[(size_t)b * DIMD + e] = acc;
  }
}

__global__ void init_cm(const float* __restrict__ c0, const float* __restrict__ m0,
                        float* __restrict__ ctrl, float* __restrict__ mem) {
  int i = blockIdx.x * blockDim.x + threadIdx.x;
  if (i < BATCH * DIMD) {
    int e = i & (DIMD - 1);
    ctrl[i] = c0[e];
    mem[i]  = m0[e];
  }
}

extern "C" void kernel_launch(void* const* d_in, const int* in_sizes, int n_in,
                              void* d_out, int out_size, void* d_ws, size_t ws_size,
                              hipStream_t stream) {
  (void)in_sizes; (void)n_in; (void)out_size; (void)ws_size;
  const float* context   = (const float*)d_in[0];
  const float* question  = (const float*)d_in[1];
  const float* knowledge = (const float*)d_in[2];
  const float* Wcq = (const float*)d_in[3];
  const float* bcq = (const float*)d_in[4];
  const float* Wca = (const float*)d_in[5];
  // d_in[6] = bca : cancels in softmax over S
  const float* Wm  = (const float*)d_in[7];
  const float* bm  = (const float*)d_in[8];
  const float* Wrc = (const float*)d_in[9];
  // d_in[10] = brc, d_in[12] = bra : cancel in softmax over N
  const float* Wra = (const float*)d_in[11];
  const float* Wwc = (const float*)d_in[13];
  const float* bwc = (const float*)d_in[14];
  const float* control0 = (const float*)d_in[15];
  const float* mem0     = (const float*)d_in[16];

  float* ws = (float*)d_ws;
  const size_t BD = (size_t)BATCH * DIMD;   // 131072
  float* cq    = ws;
  float* ctrlA = ws + 1 * BD;
  float* ctrlB = ws + 2 * BD;
  float* memA  = ws + 3 * BD;
  float* memB  = ws + 4 * BD;
  float* memp  = ws + 5 * BD;
  float* pq    = ws + 6 * BD;              // [B, 1024]
  float* readv = ws + 8 * BD;

  init_cm<<<(BATCH * DIMD + 255) / 256, 256, 0, stream>>>(control0, mem0, ctrlA, memA);

  // cq for step 0
  GemmDesc dcq0{ctrlA, question, nullptr, Wcq, bcq, cq, 1024, DIMD, 1};
  gemm_wmma_f32<<<dim3(DIMD / TN, BATCH / TM, 1), 128, 0, stream>>>(dcq0, dcq0);

  float* ctrls[2] = {ctrlA, ctrlB};
  float* mems[2]  = {memA, memB};
  for (int step = 0; step < 12; ++step) {
    float* c_new = ctrls[(step + 1) & 1];
    float* m_cur = mems[step & 1];
    float* m_new = (step == 11) ? (float*)d_out : mems[(step + 1) & 1];

    control_attn<<<BATCH, 256, 0, stream>>>(cq, context, Wca, c_new);

    // z=0: memp = mem @ Wm.T + bm   z=1: pq = (c_new * Wra) @ Wrc
    GemmDesc dmemp{m_cur, nullptr, nullptr, Wm, bm, memp, DIMD, DIMD, 1};
    GemmDesc dpq  {c_new, nullptr, Wra, Wrc, nullptr, pq, DIMD, 1024, 0};
    gemm_wmma_f32<<<dim3(1024 / TN, BATCH / TM, 2), 128, 0, stream>>>(dmemp, dpq);

    read_attn<<<BATCH, 256, 0, stream>>>(memp, pq, knowledge, readv);

    // z=0: mem' = [read|mem] @ Wwc.T + bwc   z=1: next-step cq
    GemmDesc dwr {readv, m_cur, nullptr, Wwc, bwc, m_new, 1024, DIMD, 1};
    GemmDesc dcq {c_new, question, nullptr, Wcq, bcq, cq, 1024, DIMD, 1};
    gemm_wmma_f32<<<dim3(DIMD / TN, BATCH / TM, 2), 128, 0, stream>>>(dwr, dcq);
  }
}